// PointBasedTransform_71236327571883
// MI455X (gfx1250) — compile-verified
//
#include <hip/hip_runtime.h>
#include <stdint.h>

// ---------------------------------------------------------------------------
// PointBasedTransform voxelization for MI455X (gfx1250, wave32).
// Pass 1: scatter conf-sort keys into per-voxel buckets (atomics).
// Pass 2 (fused): one wave32 per voxel --
//   * cross-lane bitonic sort of the 32-key bucket (ds lane ops),
//   * emit top-16 point ids,
//   * async global->LDS gather of the 16x29 feature tile
//     (GLOBAL_LOAD_ASYNC_TO_LDS_B32),
//   * async LDS->global drain (GLOBAL_STORE_ASYNC_FROM_LDS_B128, NT hint) --
//     the tile never passes through VGPRs.
// ---------------------------------------------------------------------------

#define GRID_G      64
#define NUM_VOXELS  (GRID_G * GRID_G * GRID_G)   // 262144
#define TOPK        16
#define FEAT        29                            // 3 + 1 + 24 + 1
#define TILE_F      (TOPK * FEAT)                 // 464 floats = 1856 B = 116*16B
#define VOX_RES     0.02f
#define MIN_CORNER  (-0.64f)
#define BUCKET_CAP  32

typedef float v4f __attribute__((ext_vector_type(4)));

// ---------------- Pass 1: scatter keys into voxel buckets -------------------
__global__ __launch_bounds__(256) void pbt_scatter(
    const float* __restrict__ pts, const float* __restrict__ conf,
    uint32_t* __restrict__ counts, uint64_t* __restrict__ buckets, int n)
{
    int i = blockIdx.x * blockDim.x + threadIdx.x;
    if (i >= n) return;
    float x = pts[3 * (size_t)i + 0];
    float y = pts[3 * (size_t)i + 1];
    float z = pts[3 * (size_t)i + 2];
    // Multiply by folded reciprocal: avoids 3 exact-division Newton chains.
    const float inv_res = 1.0f / VOX_RES;
    int ix = (int)floorf((x - MIN_CORNER) * inv_res);
    int iy = (int)floorf((y - MIN_CORNER) * inv_res);
    int iz = (int)floorf((z - MIN_CORNER) * inv_res);
    if ((unsigned)ix >= GRID_G || (unsigned)iy >= GRID_G || (unsigned)iz >= GRID_G)
        return;
    int vox = ix * (GRID_G * GRID_G) + iy * GRID_G + iz;
    // conf >= 0 => bit pattern is order-preserving; invert for descending conf.
    // Low word = point id => stable tie-break by original index ascending.
    uint32_t cb  = __float_as_uint(conf[i]);
    uint64_t key = ((uint64_t)(~cb) << 32) | (uint32_t)i;
    uint32_t pos = atomicAdd(&counts[vox], 1u);
    if (pos < BUCKET_CAP)
        buckets[(size_t)vox * BUCKET_CAP + pos] = key;
}

// ------- Pass 2 (fused): wave32 top-K select + async tile gather/drain ------
__global__ __launch_bounds__(256) void pbt_select_gather(
    const uint32_t* __restrict__ counts, const uint64_t* __restrict__ buckets,
    const float* __restrict__ pts,  const float* __restrict__ conf,
    const float* __restrict__ desc, const float* __restrict__ dconf,
    int* __restrict__ grid_out, float* __restrict__ X)
{
    // 8 waves/block, one 464-float tile each (16B-aligned, 1856 B per wave).
    __shared__ __align__(16) float stage[8][TILE_F];

    int tid  = blockIdx.x * 256 + threadIdx.x;
    int v    = tid >> 5;           // one wave per voxel
    int lane = tid & 31;
    int w    = threadIdx.x >> 5;
    if (v >= NUM_VOXELS) return;

    // Wave-uniform count -> scalar (s_load) path.
    int vs = __builtin_amdgcn_readfirstlane(v);
    uint32_t cnt = counts[vs];
    uint32_t m   = cnt < BUCKET_CAP ? cnt : BUCKET_CAP;

    uint64_t key = (lane < (int)m) ? buckets[(size_t)v * BUCKET_CAP + lane]
                                   : ~0ull;  // sentinel (unreachable by real keys)

    // Full 32-lane bitonic sort; ascending key == conf desc, index asc.
    #pragma unroll
    for (int k = 2; k <= 32; k <<= 1) {
        #pragma unroll
        for (int j = k >> 1; j > 0; j >>= 1) {
            uint64_t other =
                (uint64_t)__shfl_xor((unsigned long long)key, j, 32);
            bool     up = ((lane & k) == 0);
            uint64_t mn = key < other ? key : other;
            uint64_t mx = key < other ? other : key;
            key = (((lane & j) == 0) == up) ? mn : mx;
        }
    }

    int pid = -1;
    if (lane < TOPK) {
        pid = (key == ~0ull) ? -1 : (int)(uint32_t)(key & 0xFFFFFFFFu);
        grid_out[(size_t)v * TOPK + lane] = pid;
    }

    float* dst = X + (size_t)v * TILE_F;

    unsigned long long occ = __ballot(pid >= 0);
    if (occ == 0ull) {
        // Empty voxel: stream zeros directly, skip LDS staging.
        v4f z4 = (v4f)0.0f;
        for (int t = lane; t < TILE_F / 4; t += 32)
            __builtin_nontemporal_store(z4, (v4f*)dst + t);
        return;
    }

    // Stage the tile: zeros for empty slots (ds_store), async global->LDS
    // B32 loads for occupied components. Each element is touched exactly once.
    for (int e = lane; e < TILE_F; e += 32) {
        int slot = e / FEAT;
        int comp = e - slot * FEAT;
        int p    = __shfl(pid, slot, 32);       // broadcast slot's point id
        if (p < 0) {
            stage[w][e] = 0.0f;
        } else {
            const float* src;
            if (comp < 3)        src = pts   + 3  * (size_t)p + comp;
            else if (comp == 3)  src = conf  + p;
            else if (comp < 28)  src = desc  + 24 * (size_t)p + (comp - 4);
            else                 src = dconf + p;
            uint32_t laddr = (uint32_t)(uintptr_t)&stage[w][e]; // LDS byte addr
            uint64_t gaddr = (uint64_t)(uintptr_t)src;
            asm volatile("global_load_async_to_lds_b32 %0, %1, off"
                         :: "v"(laddr), "v"(gaddr)
                         : "memory");
        }
    }

    // LDS must be fully written before the async store engine reads it:
    // zero ds_stores (DScnt) and async fills (ASYNCcnt).
    asm volatile("s_wait_dscnt 0" ::: "memory");
    asm volatile("s_wait_asynccnt 0" ::: "memory");

    // Drain: 116 x B128 async LDS->global stores, non-temporal (486 MB stream
    // should not evict the gather working set from L2). Both LDS and global
    // chunk addresses are 16B-aligned (tile = 116*16B, bases 16B-aligned).
    uint32_t lbase = (uint32_t)(uintptr_t)&stage[w][0];
    for (int t = lane; t < TILE_F / 4; t += 32) {
        uint32_t laddr = lbase + (uint32_t)t * 16u;
        uint64_t gaddr = (uint64_t)(uintptr_t)(dst + 4 * (size_t)t);
        asm volatile("global_store_async_from_lds_b128 %0, %1, off th:TH_STORE_NT"
                     :: "v"(gaddr), "v"(laddr)
                     : "memory");
    }
    // s_endpgm performs an implicit wait-idle; outstanding async stores drain.
}

// ---------------------------------------------------------------------------
extern "C" void kernel_launch(void* const* d_in, const int* in_sizes, int n_in,
                              void* d_out, int out_size, void* d_ws, size_t ws_size,
                              hipStream_t stream)
{
    const float* pts   = (const float*)d_in[0];  // (N,3)
    const float* conf  = (const float*)d_in[1];  // (N,)
    const float* desc  = (const float*)d_in[2];  // (N,24)
    const float* dconf = (const float*)d_in[3];  // (N,)
    int n = in_sizes[1];                         // N points

    // Workspace: counts (1 MB) | buckets (64 MB)
    uint32_t* counts  = (uint32_t*)d_ws;
    uint64_t* buckets = (uint64_t*)((char*)d_ws +
                          (size_t)NUM_VOXELS * sizeof(uint32_t));

    // Outputs concatenated: pts_in_grid (int32, V*K) then X (f32, V*K*29).
    int*   grid_out = (int*)d_out;
    float* X        = (float*)d_out + (size_t)NUM_VOXELS * TOPK;

    hipMemsetAsync(counts, 0, (size_t)NUM_VOXELS * sizeof(uint32_t), stream);

    int blocksA = (n + 255) / 256;
    pbt_scatter<<<blocksA, 256, 0, stream>>>(pts, conf, counts, buckets, n);

    int blocksV = (NUM_VOXELS * 32) / 256;       // one wave32 per voxel
    pbt_select_gather<<<blocksV, 256, 0, stream>>>(
        counts, buckets, pts, conf, desc, dconf, grid_out, X);
}